// ReactionForwardPolicy_74036646248500
// MI455X (gfx1250) — compile-verified
//
#include <hip/hip_runtime.h>
#include <hip/hip_bf16.h>
#include <math.h>

typedef __attribute__((ext_vector_type(2))) float v2f;
typedef __attribute__((ext_vector_type(8))) float v8f;

#define B_STATES 2048
#define A_ACT    256
#define HID      256

// ---------------------------------------------------------------------------
// Kernel 1: hidden = gelu(all_embeddings[state_idx] @ Wb + bb)   [2048 x 256]
// One wave32 per 16x16 output tile, K-loop in steps of 4 with
// v_wmma_f32_16x16x4_f32.  A-matrix (16x4 f32) per ISA layout:
//   lanes 0-15 : M = 0..15, VGPR0 = K, VGPR1 = K+1
//   lanes 16-31: M = 0..15, VGPR0 = K+2, VGPR1 = K+3
// B-matrix (4x16) mirrored: half selects K / K+2 rows, lane%16 selects N.
// ---------------------------------------------------------------------------
__global__ __launch_bounds__(128) void mlp_b_wmma_kernel(
    const float* __restrict__ all_embeddings,
    const float* __restrict__ Wb,
    const float* __restrict__ bb,
    const int*   __restrict__ state_idx,
    float*       __restrict__ hidden)
{
    const int wave = threadIdx.x >> 5;
    const int lane = threadIdx.x & 31;
    const int tile = blockIdx.x * 4 + wave;       // 2048 tiles total
    const int m0   = (tile >> 4) << 4;            // 128 tiles along M
    const int n0   = (tile & 15) << 4;            // 16 tiles along N
    const int l    = lane & 15;
    const int half = lane >> 4;

    const float* __restrict__ arow =
        all_embeddings + (size_t)state_idx[m0 + l] * HID;

    v8f c = {};
    #pragma unroll 8
    for (int k = 0; k < HID; k += 4) {
        const int ka = k + 2 * half;              // K or K+2 per lane-half
        const float2 av = *(const float2*)(arow + ka);   // 8B-aligned pair
        v2f a, b;
        a.x = av.x;                               // K (+2*half)
        a.y = av.y;                               // K+1 (+2*half)
        b.x = Wb[(size_t)ka       * HID + n0 + l];
        b.y = Wb[(size_t)(ka + 1) * HID + n0 + l];
        // (neg_a, A, neg_b, B, c_mod, C, reuse_a, reuse_b)
        c = __builtin_amdgcn_wmma_f32_16x16x4_f32(
                false, a, false, b, (short)0, c, false, false);
    }

    // C/D layout: VGPR r -> M = r (lanes 0-15), M = r+8 (lanes 16-31); N = lane%16
    const float bias = bb[n0 + l];
    #pragma unroll
    for (int r = 0; r < 8; ++r) {
        float x = c[r] + bias;
        // exact GELU (approximate=False): 0.5*x*(1+erf(x/sqrt(2)))
        float g = 0.5f * x * (1.0f + erff(x * 0.70710678118654752f));
        hidden[(size_t)(m0 + r + 8 * half) * HID + n0 + l] = g;
    }
}

// ---------------------------------------------------------------------------
// Kernel 2: per-state ragged matvec + masked log_softmax.
// One block (8 waves) per state. Each wave streams rows a = wave, wave+8, ...
// A row (256 f32) is read as two coalesced float4 per lane; dot with
// register-resident emb fragments; 5-step shfl_xor reduction -> 1 logit/row.
// ---------------------------------------------------------------------------
__global__ __launch_bounds__(256) void score_logsoftmax_kernel(
    const float* __restrict__ b_action_table,
    const int*   __restrict__ action_idx,
    const int*   __restrict__ lengths,
    const float* __restrict__ hidden,
    float*       __restrict__ out)
{
    const int b    = blockIdx.x;
    const int t    = threadIdx.x;
    const int wave = t >> 5;
    const int lane = t & 31;

    __shared__ float s_logits[A_ACT];
    __shared__ float s_red[256];

    const float* __restrict__ emb = hidden + (size_t)b * HID;
    const float4 e0 = *(const float4*)(emb + lane * 4);        // h = 4*lane..+3
    const float4 e1 = *(const float4*)(emb + 128 + lane * 4);  // h = 128+4*lane..+3

    const int* __restrict__ aidx = action_idx + (size_t)b * A_ACT;

    for (int a = wave; a < A_ACT; a += 8) {
        const float* __restrict__ row =
            b_action_table + (size_t)aidx[a] * HID;
        if (a + 8 < A_ACT) {
            // gfx1250 global_prefetch_b8 for the next gathered row
            __builtin_prefetch(b_action_table + (size_t)aidx[a + 8] * HID + lane * 4, 0, 0);
        }
        const float4 r0 = *(const float4*)(row + lane * 4);
        const float4 r1 = *(const float4*)(row + 128 + lane * 4);
        float p = r0.x * e0.x + r0.y * e0.y + r0.z * e0.z + r0.w * e0.w
                + r1.x * e1.x + r1.y * e1.y + r1.z * e1.z + r1.w * e1.w;
        #pragma unroll
        for (int off = 16; off > 0; off >>= 1)
            p += __shfl_xor(p, off, 32);
        if (lane == 0) s_logits[a] = p;
    }
    __syncthreads();

    const int L = lengths[b];
    const float v = (t < L) ? s_logits[t] : -INFINITY;

    // block max
    s_red[t] = v;
    __syncthreads();
    #pragma unroll
    for (int s = 128; s > 0; s >>= 1) {
        if (t < s) s_red[t] = fmaxf(s_red[t], s_red[t + s]);
        __syncthreads();
    }
    const float m = s_red[0];
    __syncthreads();

    // block sum of exp
    s_red[t] = (t < L) ? __expf(v - m) : 0.0f;
    __syncthreads();
    #pragma unroll
    for (int s = 128; s > 0; s >>= 1) {
        if (t < s) s_red[t] += s_red[t + s];
        __syncthreads();
    }
    const float lse = m + __logf(s_red[0]);

    out[(size_t)b * A_ACT + t] = (t < L) ? (v - lse) : -INFINITY;
}

// ---------------------------------------------------------------------------
extern "C" void kernel_launch(void* const* d_in, const int* in_sizes, int n_in,
                              void* d_out, int out_size, void* d_ws, size_t ws_size,
                              hipStream_t stream) {
    const float* all_embeddings = (const float*)d_in[0];  // [65536, 256]
    const float* b_action_table = (const float*)d_in[1];  // [200000, 256]
    const float* Wb             = (const float*)d_in[2];  // [256, 256]
    const float* bb             = (const float*)d_in[3];  // [256]
    const int*   state_idx      = (const int*)d_in[4];    // [2048]
    const int*   action_idx     = (const int*)d_in[5];    // [2048, 256]
    const int*   lengths        = (const int*)d_in[6];    // [2048]
    float*       out            = (float*)d_out;          // [2048, 256]

    float* hidden = (float*)d_ws;                         // 2048*256*4 = 2 MiB

    // 2048 16x16 tiles, 4 waves per 128-thread block -> 512 blocks
    mlp_b_wmma_kernel<<<(B_STATES / 16) * (HID / 16) / 4, 128, 0, stream>>>(
        all_embeddings, Wb, bb, state_idx, hidden);

    score_logsoftmax_kernel<<<B_STATES, 256, 0, stream>>>(
        b_action_table, action_idx, lengths, hidden, out);
}